// Layer_84937273245883
// MI455X (gfx1250) — compile-verified
//
#include <hip/hip_runtime.h>

// MI455X / gfx1250, wave32. WMMA fp32 path: V_WMMA_F32_16X16X4_F32.
typedef __attribute__((ext_vector_type(2))) float v2f;
typedef __attribute__((ext_vector_type(8))) float v8f;
typedef __attribute__((ext_vector_type(4))) float v4f;   // clang vector: OK for NT builtin
typedef __attribute__((ext_vector_type(4))) int   v4i;

#define N2_ 512
#define T_  4096
#define E_  512
#define D_  256

// ---------------------------------------------------------------------------
// Step 1: new2[j,d] = sum_i x2[i,d]*W2[j,i,d] + sum_i R2[j,i,d] + x2[j,d]
// 512 MB streamed once from HBM -> dominates runtime. B128 loads, and
// NON-TEMPORAL hint so the one-shot W2/R2 stream doesn't evict the L2-resident
// data (A2/A3/spec/new_common) that the WMMA GEMMs re-read 16-32x.
// ---------------------------------------------------------------------------
__global__ void step1_new2_kernel(const float* __restrict__ emb,
                                  const float* __restrict__ W2,
                                  const float* __restrict__ R2,
                                  float* __restrict__ new2) {
    __shared__ v4f sm[256];
    const int j    = blockIdx.x;
    const int tid  = threadIdx.x;
    const int d4   = (tid & 63) << 2;   // 0..252 step 4
    const int isub = tid >> 6;          // 0..3
    v4f acc = {0.f, 0.f, 0.f, 0.f};
    const size_t base = (size_t)j * N2_ * D_;
    for (int i = isub; i < N2_; i += 4) {
        const v4f w = __builtin_nontemporal_load(
            (const v4f*)(W2 + base + (size_t)i * D_ + d4));
        const v4f r = __builtin_nontemporal_load(
            (const v4f*)(R2 + base + (size_t)i * D_ + d4));
        const v4f x = *(const v4f*)(emb + (size_t)i * D_ + d4);  // L2-resident
        acc += x * w + r;
    }
    sm[tid] = acc;
    __syncthreads();
    if (isub == 0) {
        const v4f x = *(const v4f*)(emb + (size_t)j * D_ + d4);
        const v4f o = sm[tid] + sm[tid + 64] + sm[tid + 128] + sm[tid + 192] + x;
        *(v4f*)(new2 + (size_t)j * D_ + d4) = o;
    }
}

__global__ void zero_f32(float* __restrict__ p, int n) {
    const int i = blockIdx.x * blockDim.x + threadIdx.x;
    if (i < n) p[i] = 0.f;
}

// S[d] = sum over t of emb[512+t, d]; 64 blocks x 64 rows, atomic combine.
__global__ void col_sum_S(const float* __restrict__ emb, float* __restrict__ S) {
    const int d  = threadIdx.x;
    const int r0 = N2_ + blockIdx.x * 64;
    float s = 0.f;
    for (int r = r0; r < r0 + 64; ++r) s += emb[(size_t)r * D_ + d];
    atomicAdd(&S[d], s);
}

// deg1[t] = 1 + nnz(A1 row t); one block per row, LDS tree reduce.
// A1 is 64 MB read exactly once -> non-temporal (B128 over v4i).
__global__ void row_count_a1(const int* __restrict__ A1, float* __restrict__ deg1) {
    __shared__ float sm[256];
    const int t = blockIdx.x;
    const v4i* row = (const v4i*)(A1 + (size_t)t * T_);
    float cnt = 0.f;
    for (int i = threadIdx.x; i < T_ / 4; i += 256) {
        const v4i v = __builtin_nontemporal_load(row + i);
        cnt += ((v.x != 0) ? 1.f : 0.f) + ((v.y != 0) ? 1.f : 0.f) +
               ((v.z != 0) ? 1.f : 0.f) + ((v.w != 0) ? 1.f : 0.f);
    }
    sm[threadIdx.x] = cnt;
    __syncthreads();
    for (int s = 128; s > 0; s >>= 1) {
        if (threadIdx.x < s) sm[threadIdx.x] += sm[threadIdx.x + s];
        __syncthreads();
    }
    if (threadIdx.x == 0) deg1[t] = 1.f + sm[0];
}

// Column nonzero counts of an RxC int matrix; grid.y chunks rows by 64.
// A2/A3 are re-read by the GEMMs -> keep regular-temporal (warms L2).
__global__ void col_count(const int* __restrict__ A, float* __restrict__ deg, int C) {
    const int c  = blockIdx.x * blockDim.x + threadIdx.x;
    const int r0 = blockIdx.y * 64;
    float cnt = 0.f;
    for (int r = r0; r < r0 + 64; ++r)
        cnt += (A[(size_t)r * C + c] != 0) ? 1.f : 0.f;
    atomicAdd(&deg[c], cnt);
}

// new1[t,d] = (sub[t,d] + S[d]) * (1 - S[d]/deg1[t])
__global__ void new1_kernel(const float* __restrict__ emb, const float* __restrict__ S,
                            const float* __restrict__ deg1, float* __restrict__ new1) {
    const int t = blockIdx.x, d = threadIdx.x;
    const float s   = S[d];
    const float sub = emb[(size_t)(N2_ + t) * D_ + d];
    new1[(size_t)t * D_ + d] = (sub + s) * (1.f - s / deg1[t]);
}

// ---------------------------------------------------------------------------
// GEMM1 (WMMA f32 16x16x4): out[512+t, d] = new1[t,d]
//   + sum_e (A2[e,t]!=0) * new2[e,d] + (512 - deg2[t])
// M=t(4096), N=d(256), K=e(512). One wave -> one 16x16 C tile, 8 waves/block.
// Two independent accumulators (even/odd K-step) to break the WMMA RAW chain;
// C/D layout is elementwise-linear, so acc0+acc1 is a valid final C.
// ---------------------------------------------------------------------------
__global__ void gemm_msg2(const int* __restrict__ A2, const float* __restrict__ spec,
                          const float* __restrict__ new1, const float* __restrict__ deg2,
                          float* __restrict__ out) {
    const int wave = threadIdx.x >> 5;
    const int lane = threadIdx.x & 31;
    const int tile = blockIdx.x * 8 + wave;   // 4096 tiles
    const int tM   = tile >> 4;               // 0..255 (t)
    const int tN   = tile & 15;               // 0..15  (d)
    const int half = lane >> 4;
    const int l    = lane & 15;
    const int tRow = tM * 16 + l;             // A-matrix M index (t)
    const int dCol = tN * 16 + l;             // B/C N index (d)
    v8f acc0 = {}, acc1 = {};
    for (int k0 = 0; k0 < E_; k0 += 8) {
        const int ka = k0 + half * 2;
        v2f a0, b0, a1, b1;
        a0.x = (A2[(size_t)(ka + 0) * T_ + tRow] != 0) ? 1.f : 0.f;
        a0.y = (A2[(size_t)(ka + 1) * T_ + tRow] != 0) ? 1.f : 0.f;
        a1.x = (A2[(size_t)(ka + 4) * T_ + tRow] != 0) ? 1.f : 0.f;
        a1.y = (A2[(size_t)(ka + 5) * T_ + tRow] != 0) ? 1.f : 0.f;
        b0.x = spec[(size_t)(ka + 0) * D_ + dCol];
        b0.y = spec[(size_t)(ka + 1) * D_ + dCol];
        b1.x = spec[(size_t)(ka + 4) * D_ + dCol];
        b1.y = spec[(size_t)(ka + 5) * D_ + dCol];
        acc0 = __builtin_amdgcn_wmma_f32_16x16x4_f32(false, a0, false, b0,
                                                     (short)0, acc0, false, false);
        acc1 = __builtin_amdgcn_wmma_f32_16x16x4_f32(false, a1, false, b1,
                                                     (short)0, acc1, false, false);
    }
#pragma unroll
    for (int r = 0; r < 8; ++r) {
        const int m = r + half * 8;
        const int t = tM * 16 + m;
        const float v = acc0[r] + acc1[r] + new1[(size_t)t * D_ + dCol]
                        + (512.f - deg2[t]);
        out[(size_t)(N2_ + t) * D_ + dCol] = v;   // new_common
    }
}

// ---------------------------------------------------------------------------
// GEMM2 (WMMA f32 16x16x4): out[e,d] = new2[e,d] *
//   (1 - (sum_t (A3[t,e]!=0)*new_common[t,d] + (4096 - deg3[e])) / (1 + deg3[e]))
// M=e(512), N=d(256), K=t(4096). Reads new_common back from d_out rows 512..
// ---------------------------------------------------------------------------
__global__ void gemm_msg3(const int* __restrict__ A3, const float* __restrict__ new2,
                          const float* __restrict__ deg3, float* __restrict__ out) {
    const int wave = threadIdx.x >> 5;
    const int lane = threadIdx.x & 31;
    const int tile = blockIdx.x * 8 + wave;   // 512 tiles
    const int tM   = tile >> 4;               // 0..31 (e)
    const int tN   = tile & 15;               // 0..15 (d)
    const int half = lane >> 4;
    const int l    = lane & 15;
    const int eRow = tM * 16 + l;
    const int dCol = tN * 16 + l;
    const float* nc = out + (size_t)N2_ * D_; // new_common, written by gemm_msg2
    v8f acc0 = {}, acc1 = {};
    for (int k0 = 0; k0 < T_; k0 += 8) {
        const int ka = k0 + half * 2;
        v2f a0, b0, a1, b1;
        a0.x = (A3[(size_t)(ka + 0) * E_ + eRow] != 0) ? 1.f : 0.f;
        a0.y = (A3[(size_t)(ka + 1) * E_ + eRow] != 0) ? 1.f : 0.f;
        a1.x = (A3[(size_t)(ka + 4) * E_ + eRow] != 0) ? 1.f : 0.f;
        a1.y = (A3[(size_t)(ka + 5) * E_ + eRow] != 0) ? 1.f : 0.f;
        b0.x = nc[(size_t)(ka + 0) * D_ + dCol];
        b0.y = nc[(size_t)(ka + 1) * D_ + dCol];
        b1.x = nc[(size_t)(ka + 4) * D_ + dCol];
        b1.y = nc[(size_t)(ka + 5) * D_ + dCol];
        acc0 = __builtin_amdgcn_wmma_f32_16x16x4_f32(false, a0, false, b0,
                                                     (short)0, acc0, false, false);
        acc1 = __builtin_amdgcn_wmma_f32_16x16x4_f32(false, a1, false, b1,
                                                     (short)0, acc1, false, false);
    }
#pragma unroll
    for (int r = 0; r < 8; ++r) {
        const int m = r + half * 8;
        const int e = tM * 16 + m;
        const float dg = deg3[e];
        const float v = new2[(size_t)e * D_ + dCol] *
                        (1.f - (acc0[r] + acc1[r] + (4096.f - dg)) / (1.f + dg));
        out[(size_t)e * D_ + dCol] = v;       // new_spec (rows 0..511; no overlap with nc)
    }
}

extern "C" void kernel_launch(void* const* d_in, const int* in_sizes, int n_in,
                              void* d_out, int out_size, void* d_ws, size_t ws_size,
                              hipStream_t stream) {
    const float* emb = (const float*)d_in[0];   // (4608, 256) f32
    const float* W2  = (const float*)d_in[1];   // (512, 512, 256) f32
    const float* R2  = (const float*)d_in[2];   // (512, 512, 256) f32
    const int*   A1  = (const int*)d_in[3];     // (4096, 4096) i32
    const int*   A2  = (const int*)d_in[4];     // (512, 4096) i32
    const int*   A3  = (const int*)d_in[5];     // (4096, 512) i32
    float* out = (float*)d_out;                 // (4608, 256) f32

    // Workspace layout (floats): new2 | new1 | S | deg1 | deg2 | deg3  (~4.6 MB)
    float* ws   = (float*)d_ws;
    float* new2 = ws;                            // 512*256
    float* new1 = ws + 131072;                   // 4096*256
    float* S    = ws + 131072 + 1048576;         // 256
    float* deg1 = S + 256;                       // 4096
    float* deg2 = deg1 + 4096;                   // 4096
    float* deg3 = deg2 + 4096;                   // 512

    // Zero atomic targets (S, deg1, deg2, deg3 region = 8960 floats).
    zero_f32<<<35, 256, 0, stream>>>(S, 256 + 4096 + 4096 + 512);

    // Step 1 (bandwidth dominant: 512 MB of W2+R2, non-temporal).
    step1_new2_kernel<<<512, 256, 0, stream>>>(emb, W2, R2, new2);

    // Reductions (independent of step 1).
    col_sum_S<<<64, 256, 0, stream>>>(emb, S);
    row_count_a1<<<4096, 256, 0, stream>>>(A1, deg1);
    col_count<<<dim3(16, 8), 256, 0, stream>>>(A2, deg2, 4096);
    col_count<<<dim3(2, 64), 256, 0, stream>>>(A3, deg3, 512);

    // Step 2 elementwise.
    new1_kernel<<<4096, 256, 0, stream>>>(emb, S, deg1, new1);

    // Step 3: WMMA GEMM + epilogue -> out rows [512, 4608).
    gemm_msg2<<<512, 256, 0, stream>>>(A2, new2, new1, deg2, out);

    // Step 4: WMMA GEMM + epilogue -> out rows [0, 512).
    gemm_msg3<<<64, 256, 0, stream>>>(A3, new2, deg3, out);
}